// Spacial_IRNN_42417097016087
// MI455X (gfx1250) — compile-verified
//
#include <hip/hip_runtime.h>
#include <hip/hip_bf16.h>

#define B_ 4
#define C_ 128
#define H_ 256
#define W_ 256
#define NPLANE ((size_t)B_ * C_ * H_ * W_)

// ---- CDNA5 feature probes (compile-time) -----------------------------------
#if defined(__has_builtin)
#if __has_builtin(__builtin_amdgcn_global_load_async_to_lds_b32)
#define HAVE_ASYNC_LDS32 1
#endif
#if __has_builtin(__builtin_amdgcn_global_load_async_to_lds_b128)
#define HAVE_ASYNC_LDS128 1
#endif
#if __has_builtin(__builtin_amdgcn_global_store_async_from_lds_b128)
#define HAVE_ASYNC_ST128 1
#endif
#if __has_builtin(__builtin_amdgcn_s_wait_asynccnt)
#define HAVE_WAIT_ASYNC 1
#endif
#endif
#ifndef HAVE_ASYNC_LDS32
#define HAVE_ASYNC_LDS32 0
#endif
#ifndef HAVE_ASYNC_LDS128
#define HAVE_ASYNC_LDS128 0
#endif
#ifndef HAVE_ASYNC_ST128
#define HAVE_ASYNC_ST128 0
#endif
#ifndef HAVE_WAIT_ASYNC
#define HAVE_WAIT_ASYNC 0
#endif

typedef int v4i __attribute__((ext_vector_type(4)));
typedef __attribute__((address_space(1))) int  gint_t;
typedef __attribute__((address_space(3))) int  lint_t;
typedef __attribute__((address_space(1))) v4i  gv4i_t;
typedef __attribute__((address_space(3))) v4i  lv4i_t;

__device__ __forceinline__ void wait_async0() {
#if HAVE_WAIT_ASYNC
  __builtin_amdgcn_s_wait_asynccnt(0);
#else
  asm volatile("s_wait_asynccnt 0" ::: "memory");
#endif
}

// ---------------------------------------------------------------------------
// Vertical scans (up / down): thread <-> (b, c, w) column; scan along H.
// Per step, a wave touches one contiguous 128B line -> fully coalesced.
// ---------------------------------------------------------------------------
__global__ __launch_bounds__(256) void irnn_vscan(
    const float* __restrict__ x, float* __restrict__ out,
    const float* __restrict__ wgt, const float* __restrict__ bs, int reverse)
{
  const int t = blockIdx.x * 256 + threadIdx.x;      // 0 .. B*C*W-1
  const int w = t % W_;
  const int bc = t / W_;                             // b*C + c
  const int c = bc % C_;
  const float wc = wgt[c];
  const float bcst = bs[c];

  const long base = (long)bc * (H_ * W_) + w;
  const long step = reverse ? -(long)W_ : (long)W_;
  long off = reverse ? base + (long)(H_ - 1) * W_ : base;

  float prev = x[off];                               // boundary row: passthrough
  out[off] = prev;
  off += step;

#pragma unroll 4
  for (int h = 1; h < H_; ++h, off += step) {
    if (h + 8 < H_) __builtin_prefetch((const void*)(x + off + 8 * step), 0, 0);
    const float xv = x[off];
    const float o = fmaxf(0.0f, fmaf(wc, prev, bcst + xv));
    out[off] = o;
    prev = o;
  }
}

// ---------------------------------------------------------------------------
// Horizontal scans (right / left): thread <-> row (b, c, h); scan along W.
// Stage 256 rows x 32 cols through LDS so global traffic stays coalesced.
// Stage-in uses async global->LDS (b128 preferred), stage-out uses async
// LDS->global stores when available; all tracked with ASYNCcnt.
// Row stride 36 floats = 144B: 16B-aligned for b128, mild 2-way bank conflict.
// ---------------------------------------------------------------------------
#define HROWS 256
#define HCH   32
#define HPAD  36

__global__ __launch_bounds__(256) void irnn_hscan(
    const float* __restrict__ x, float* __restrict__ out,
    const float* __restrict__ wgt, const float* __restrict__ bs, int reverse)
{
  __shared__ __align__(16) float tile[HROWS][HPAD];

  const int tid = threadIdx.x;
  const long r0 = (long)blockIdx.x * HROWS;          // first global row
  const int c = (int)((r0 + tid) >> 8) & (C_ - 1);   // H_=256, C_=128
  const float wc = wgt[c];
  const float bcst = bs[c];

  float prev = 0.0f;

  for (int k = 0; k < W_ / HCH; ++k) {
    const int cb = (reverse ? (W_ / HCH - 1 - k) : k) * HCH;

    // ---- stage in (coalesced; async-to-LDS, widest available) ----
#if HAVE_ASYNC_LDS128
    // 2048 float4's; 8 per thread
    for (int i = tid; i < HROWS * (HCH / 4); i += 256) {
      const int row = i >> 3;
      const int col4 = (i & 7) * 4;
      const long g = (r0 + row) * W_ + cb + col4;
      __builtin_amdgcn_global_load_async_to_lds_b128(
          (gv4i_t*)(x + g), (lv4i_t*)&tile[row][col4], 0, 0);
    }
    wait_async0();
#elif HAVE_ASYNC_LDS32
    for (int i = tid; i < HROWS * HCH; i += 256) {
      const int row = i >> 5;
      const int col = i & (HCH - 1);
      const long g = (r0 + row) * W_ + cb + col;
      __builtin_amdgcn_global_load_async_to_lds_b32(
          (gint_t*)(x + g), (lint_t*)&tile[row][col], 0, 0);
    }
    wait_async0();
#else
    for (int i = tid; i < HROWS * (HCH / 4); i += 256) {
      const int row = i >> 3;
      const int col4 = (i & 7) * 4;
      const long g = (r0 + row) * W_ + cb + col4;
      *(float4*)&tile[row][col4] = *(const float4*)(x + g);
    }
#endif
    __syncthreads();

    // ---- sequential scan over this chunk, in place in LDS ----
#pragma unroll
    for (int j = 0; j < HCH; ++j) {
      const int col = reverse ? (HCH - 1 - j) : j;
      const float xv = tile[tid][col];
      float o;
      if (k == 0 && j == 0)
        o = xv;                                      // boundary col: passthrough
      else
        o = fmaxf(0.0f, fmaf(wc, prev, bcst + xv));
      tile[tid][col] = o;                            // safe: own row only
      prev = o;
    }
    __syncthreads();

    // ---- stage out (coalesced; async LDS->global when available) ----
#if HAVE_ASYNC_ST128
    for (int i = tid; i < HROWS * (HCH / 4); i += 256) {
      const int row = i >> 3;
      const int col4 = (i & 7) * 4;
      const long g = (r0 + row) * W_ + cb + col4;
      __builtin_amdgcn_global_store_async_from_lds_b128(
          (gv4i_t*)(out + g), (lv4i_t*)&tile[row][col4], 0, 0);
    }
    // Drain before next chunk's async loads may overwrite this LDS tile.
    wait_async0();
#else
    for (int i = tid; i < HROWS * (HCH / 4); i += 256) {
      const int row = i >> 3;
      const int col4 = (i & 7) * 4;
      const long g = (r0 + row) * W_ + cb + col4;
      *(float4*)(out + g) = *(const float4*)&tile[row][col4];
    }
#endif
    __syncthreads();
  }
}

// ---------------------------------------------------------------------------
extern "C" void kernel_launch(void* const* d_in, const int* in_sizes, int n_in,
                              void* d_out, int out_size, void* d_ws, size_t ws_size,
                              hipStream_t stream) {
  (void)in_sizes; (void)n_in; (void)out_size; (void)d_ws; (void)ws_size;

  const float* x    = (const float*)d_in[0];
  const float* w_up = (const float*)d_in[1];
  const float* b_up = (const float*)d_in[2];
  const float* w_ri = (const float*)d_in[3];
  const float* b_ri = (const float*)d_in[4];
  const float* w_dn = (const float*)d_in[5];
  const float* b_dn = (const float*)d_in[6];
  const float* w_le = (const float*)d_in[7];
  const float* b_le = (const float*)d_in[8];
  float* out = (float*)d_out;

  const dim3 blk(256);
  const dim3 gv((B_ * C_ * W_) / 256);   // 512 blocks
  const dim3 gh((B_ * C_ * H_) / 256);   // 512 blocks

  // Output order: (up, right, down, left)
  irnn_vscan<<<gv, blk, 0, stream>>>(x, out + 0 * NPLANE, w_up, b_up, /*reverse=*/1);
  irnn_hscan<<<gh, blk, 0, stream>>>(x, out + 1 * NPLANE, w_ri, b_ri, /*reverse=*/0);
  irnn_vscan<<<gv, blk, 0, stream>>>(x, out + 2 * NPLANE, w_dn, b_dn, /*reverse=*/0);
  irnn_hscan<<<gh, blk, 0, stream>>>(x, out + 3 * NPLANE, w_le, b_le, /*reverse=*/1);
}